// HF_43250320670768
// MI455X (gfx1250) — compile-verified
//
#include <hip/hip_runtime.h>

#ifndef __has_builtin
#define __has_builtin(x) 0
#endif

#if __has_builtin(__builtin_amdgcn_global_load_async_to_lds_b128)
#define USE_ASYNC_LDS 1
#else
#define USE_ASYNC_LDS 0
#endif

typedef float v4f __attribute__((ext_vector_type(4)));
typedef int   v4i __attribute__((ext_vector_type(4)));

// Pointer flavors for the CDNA5 async global->LDS builtin:
// param0: v4i in addrspace(1) (global), param1: v4i in addrspace(3) (LDS).
typedef __attribute__((address_space(1))) v4i* gptr_v4i;
typedef __attribute__((address_space(3))) v4i* lptr_v4i;

#define ROW_LEN 1024   // L from the reference
#define TPB     256    // 8 wave32s; each thread owns 4 consecutive floats

static __device__ __forceinline__ void wait_asynccnt0() {
#if __has_builtin(__builtin_amdgcn_s_wait_asynccnt)
  __builtin_amdgcn_s_wait_asynccnt(0);
#else
  asm volatile("s_wait_asynccnt 0" ::: "memory");
#endif
}

// One workgroup per row. Row chunk layout: thread t covers elements
// [4t, 4t+4) -> lane-contiguous 16B -> coalesced b128 transactions.
__global__ __launch_bounds__(TPB) void hh_rowreflect(
    const float* __restrict__ v,
    const float* __restrict__ z,
    float* __restrict__ out,
    int nrows)
{
#if USE_ASYNC_LDS
  __shared__ float s_v[ROW_LEN];   // 4 KiB staging for the v row (async path)
#endif
  __shared__ float s_part[16];     // 8 wave partials for vz + 8 for vv

  const int row = blockIdx.x;
  if (row >= nrows) return;
  const int    t    = threadIdx.x;
  const size_t base = (size_t)row * ROW_LEN + (size_t)(t << 2);

#if USE_ASYNC_LDS
  // CDNA5 async global->LDS copy of this lane's 16 bytes of v (ASYNCcnt path).
  __builtin_amdgcn_global_load_async_to_lds_b128(
      (gptr_v4i)(v + base),
      (lptr_v4i)(&s_v[t << 2]),
      0, 0);
#endif

  // z chunk straight to VGPRs, non-temporal (stream-once data).
  const v4f z4 = __builtin_nontemporal_load((const v4f*)(z + base));

#if USE_ASYNC_LDS
  wait_asynccnt0();                 // drain this wave's async copies
  __syncthreads();                  // all waves' LDS data resident
  const v4f v4 = *(const v4f*)(&s_v[t << 2]);   // ds_load_b128
#else
  const v4f v4 = __builtin_nontemporal_load((const v4f*)(v + base));
#endif

  // Per-thread partial dot products.
  float vz = v4.x * z4.x + v4.y * z4.y + v4.z * z4.z + v4.w * z4.w;
  float vv = v4.x * v4.x + v4.y * v4.y + v4.z * v4.z + v4.w * v4.w;

  // wave32 butterfly reduction (5 levels).
  #pragma unroll
  for (int m = 16; m >= 1; m >>= 1) {
    vz += __shfl_xor(vz, m, 32);
    vv += __shfl_xor(vv, m, 32);
  }

  // Cross-wave reduction through 16 floats of LDS.
  const int wave = t >> 5;
  if ((t & 31) == 0) {
    s_part[wave]     = vz;
    s_part[8 + wave] = vv;
  }
  __syncthreads();

  float tvz = 0.f, tvv = 0.f;
  #pragma unroll
  for (int w = 0; w < 8; ++w) {
    tvz += s_part[w];
    tvv += s_part[8 + w];
  }

  const float scale = 2.0f * tvz / tvv;   // matches reference: z - 2 v (v.z)/||v||^2

  v4f o;
  o.x = z4.x - scale * v4.x;
  o.y = z4.y - scale * v4.y;
  o.z = z4.z - scale * v4.z;
  o.w = z4.w - scale * v4.w;

  // Write-once output: non-temporal b128 store, keep L2 clean.
  __builtin_nontemporal_store(o, (v4f*)(out + base));
}

extern "C" void kernel_launch(void* const* d_in, const int* in_sizes, int n_in,
                              void* d_out, int out_size, void* d_ws, size_t ws_size,
                              hipStream_t stream) {
  (void)n_in; (void)d_ws; (void)ws_size; (void)out_size;
  const float* v = (const float*)d_in[0];
  const float* z = (const float*)d_in[1];
  float* out = (float*)d_out;
  const int nrows = in_sizes[0] / ROW_LEN;   // 16384 for the reference shapes

  hipLaunchKernelGGL(hh_rowreflect, dim3(nrows), dim3(TPB), 0, stream,
                     v, z, out, nrows);
}